// TensorfEncoder_10874857193775
// MI455X (gfx1250) — compile-verified
//
#include <hip/hip_runtime.h>

typedef __attribute__((ext_vector_type(2))) float v2f;
typedef __attribute__((ext_vector_type(4))) float v4f;
typedef __attribute__((ext_vector_type(8))) float v8f;

#define CCH   8      // channels per plane
#define RES   300
#define APP   32     // output features
#define KF    24     // 3*CCH feature length
#define BLK   256
#define OSTRIDE 36   // outstage row stride (floats): 8*36 % 64 == 32 -> conflict-free halves

// MAT_MODE = [[0,1],[0,2],[1,2]], VEC_MODE = [2,1,0]
__device__ __constant__ int MAT0_c[3] = {0, 0, 1};
__device__ __constant__ int MAT1_c[3] = {1, 2, 2};
__device__ __constant__ int VECM_c[3] = {2, 1, 0};

__global__ __launch_bounds__(BLK) void tensorf_encoder_kernel(
    const float* __restrict__ inputs,   // [N,3]
    const float* __restrict__ planes,   // [3,C,RES,RES]
    const float* __restrict__ Wm,       // [APP, KF]
    const int*   __restrict__ size_p,   // scalar
    float*       __restrict__ out,      // [N, APP]
    int N)
{
    __shared__ float feat[BLK][KF + 1];           // stride 25 -> bank-conflict free
    __shared__ float outstage[8][16 * OSTRIDE];   // per-wave 16x32 D staging, padded

    const int tid = threadIdx.x;
    const int pid_raw = blockIdx.x * BLK + tid;
    const int pid = pid_raw < N ? pid_raw : (N - 1);   // clamped load, no divergence

    const float inv = 1.0f / (float)size_p[0];
    float c0 = inputs[pid * 3 + 0] * inv;
    float c1 = inputs[pid * 3 + 1] * inv;
    float c2 = inputs[pid * 3 + 2] * inv;
    float c[3] = {c0, c1, c2};

    // ---------------- Phase 1: 24 features per point -> LDS ----------------
    #pragma unroll
    for (int i = 0; i < 3; ++i) {
        const float gx = c[MAT0_c[i]];
        const float gy = c[MAT1_c[i]];
        const float gv = c[VECM_c[i]];

        // align_corners=True mapping: x = (g+1)*0.5*(RES-1)
        const float x  = fmaf(gx, 149.5f, 149.5f);
        const float y  = fmaf(gy, 149.5f, 149.5f);
        const float yl = fmaf(gv, 149.5f, 149.5f);

        float xf  = fminf(fmaxf(floorf(x),  0.0f), (float)(RES - 2));
        float yf  = fminf(fmaxf(floorf(y),  0.0f), (float)(RES - 2));
        float ylf = fminf(fmaxf(floorf(yl), 0.0f), (float)(RES - 2));
        const int x0  = (int)xf;
        const int y0  = (int)yf;
        const int yl0 = (int)ylf;
        const float wx = x - xf;
        const float wy = y - yf;
        const float wv = yl - ylf;

        const float w00 = (1.0f - wx) * (1.0f - wy);
        const float w01 = wx * (1.0f - wy);
        const float w10 = (1.0f - wx) * wy;
        const float w11 = wx * wy;

        const float* pl = planes + (size_t)i * CCH * RES * RES + y0 * RES + x0;
        // line sample: gx=0 -> x=149.5 exactly, wx=0.5 folded into 0.5*(a+b)
        const float* ll = planes + (size_t)i * CCH * RES * RES + yl0 * RES + 149;

        #pragma unroll
        for (int ch = 0; ch < CCH; ++ch) {
            const float* b = pl + (size_t)ch * RES * RES;
            const float p = b[0] * w00 + b[1] * w01 + b[RES] * w10 + b[RES + 1] * w11;

            const float* lb = ll + (size_t)ch * RES * RES;
            const float l0 = 0.5f * (lb[0] + lb[1]);
            const float l1 = 0.5f * (lb[RES] + lb[RES + 1]);
            const float lv = fmaf(l1 - l0, wv, l0);   // l0*(1-wv) + l1*wv

            feat[tid][i * CCH + ch] = p * lv;
        }
    }

    __syncthreads();

    // ---------------- Phase 2: WMMA f32 16x16x4, K=24 in 6 chunks ----------
    const int lane = tid & 31;
    const int wave = tid >> 5;
    const int n    = lane & 15;   // A: row m; B/D: col n
    const int h    = lane >> 4;   // K half-select

    // B fragments: B[k][ncol] = W[ncol][k]; lane holds k = 4*kc + 2*h + {0,1}
    v2f bfrag[2][6];
    #pragma unroll
    for (int tn = 0; tn < 2; ++tn) {
        #pragma unroll
        for (int kc = 0; kc < 6; ++kc) {
            const int kk  = kc * 4 + 2 * h;
            const int col = tn * 16 + n;
            bfrag[tn][kc].x = Wm[col * KF + kk];
            bfrag[tn][kc].y = Wm[col * KF + kk + 1];
        }
    }

    // Uniform (scalar) predicate: whole block's outputs in range?
    const bool full = ((unsigned)(blockIdx.x + 1) * BLK) <= (unsigned)N;
    float* stg = &outstage[wave][0];

    #pragma unroll
    for (int t = 0; t < 2; ++t) {
        const int rowbase = wave * 32 + t * 16;   // this wave's 16-point tile in LDS
        v8f d0 = {};
        v8f d1 = {};
        #pragma unroll
        for (int kc = 0; kc < 6; ++kc) {
            const int kk = kc * 4 + 2 * h;
            v2f a;
            a.x = feat[rowbase + n][kk];
            a.y = feat[rowbase + n][kk + 1];
            d0 = __builtin_amdgcn_wmma_f32_16x16x4_f32(
                    false, a, false, bfrag[0][kc], (short)0, d0, false, false);
            d1 = __builtin_amdgcn_wmma_f32_16x16x4_f32(
                    false, a, false, bfrag[1][kc], (short)0, d1, false, false);
        }

        const int prow = blockIdx.x * BLK + rowbase;

        if (full) {
            // Stage D tile (16 rows x 32 cols) into LDS, then coalesced b128 stores.
            // D layout: VGPR r -> row m = r + 8*h, col n (d0: cols 0-15, d1: 16-31)
            #pragma unroll
            for (int r = 0; r < 8; ++r) {
                const int m = r + 8 * h;
                stg[m * OSTRIDE + n]      = d0[r];
                stg[m * OSTRIDE + 16 + n] = d1[r];
            }
            // Intra-wave RAW on LDS: in-order DS + compiler-inserted s_wait_dscnt.
            #pragma unroll
            for (int j = 0; j < 4; ++j) {
                const int cid = lane + 32 * j;        // float4 chunk id in 16x32 tile
                const int row = cid >> 3;             // 8 chunks per row
                const int col = (cid & 7) * 4;
                v4f v = *(const v4f*)&stg[row * OSTRIDE + col];      // 16B aligned
                *(v4f*)&out[(size_t)(prow + row) * APP + col] = v;   // 16B aligned
            }
        } else {
            // Rare tail path: guarded scalar stores.
            #pragma unroll
            for (int r = 0; r < 8; ++r) {
                const int m = r + 8 * h;
                const int gr = prow + m;
                if (gr < N) {
                    out[(size_t)gr * APP + n]      = d0[r];
                    out[(size_t)gr * APP + 16 + n] = d1[r];
                }
            }
        }
    }
}

extern "C" void kernel_launch(void* const* d_in, const int* in_sizes, int n_in,
                              void* d_out, int out_size, void* d_ws, size_t ws_size,
                              hipStream_t stream) {
    const float* inputs = (const float*)d_in[0];   // [N,3] f32
    const float* planes = (const float*)d_in[1];   // [3,8,300,300] f32
    const float* Wm     = (const float*)d_in[2];   // [32,24] f32
    const int*   size_p = (const int*)d_in[3];     // scalar int

    float* out = (float*)d_out;
    const int N = in_sizes[0] / 3;
    const int blocks = (N + BLK - 1) / BLK;

    tensorf_encoder_kernel<<<blocks, BLK, 0, stream>>>(inputs, planes, Wm, size_p, out, N);
}